// SchNetInteraction_89713276879260
// MI455X (gfx1250) — compile-verified
//
#include <hip/hip_runtime.h>
#include <hip/hip_bf16.h>

typedef __attribute__((ext_vector_type(2))) float v2f;
typedef __attribute__((ext_vector_type(8))) float v8f;

#define NF    128   // filters / basis
#define NRBF  20    // radial basis
#define PITCH 132   // LDS row pitch (floats) -> bank-conflict-free
#define SUBT  4     // 16-edge sub-tiles per block (64 edges/block)

__device__ __forceinline__ float ssp(float x) {
  // shifted softplus, numerically stable: max(x,0)+log1p(exp(-|x|)) - log(2)
  return fmaxf(x, 0.0f) + log1pf(__expf(-fabsf(x))) - 0.69314718055994531f;
}

// ---- Wave-level 16x16 f32 WMMA tile GEMM, B from global (W^T chunk) ----
// A fragment (16x4 f32): lane<16 holds M=lane, K=k+{0,1}; lane>=16 M=lane-16, K=k+{2,3}.
// B fragment mirrors:    lane holds N=lane%16,  K=k+2*(lane/16)+{0,1}.
template<int K, int SA>
__device__ __forceinline__ v8f wmma_f32_acc(const float* __restrict__ As,
                                            const float* __restrict__ W, int ldW,
                                            int nbase, v8f c, int lane) {
  const int l16  = lane & 15;
  const int koff = (lane >> 4) << 1;
  const float* arow = As + l16 * SA + koff;
  const float* wrow = W + (size_t)(nbase + l16) * ldW + koff;
#pragma unroll
  for (int k = 0; k < K; k += 4) {
    v2f a, b;
    a.x = arow[k]; a.y = arow[k + 1];
    b.x = wrow[k]; b.y = wrow[k + 1];
    c = __builtin_amdgcn_wmma_f32_16x16x4_f32(false, a, false, b, (short)0, c, false, false);
  }
  return c;
}

// ---- Same, but B fragments pre-loaded in registers (edge-invariant weights) ----
template<int NIT, int SA>
__device__ __forceinline__ v8f wmma_f32_accB(const float* __restrict__ As,
                                             const v2f* __restrict__ B,
                                             v8f c, int lane) {
  const int l16  = lane & 15;
  const int koff = (lane >> 4) << 1;
  const float* arow = As + l16 * SA + koff;
#pragma unroll
  for (int k = 0; k < NIT; ++k) {
    v2f a;
    a.x = arow[4 * k]; a.y = arow[4 * k + 1];
    c = __builtin_amdgcn_wmma_f32_16x16x4_f32(false, a, false, B[k], (short)0, c, false, false);
  }
  return c;
}

__global__ void k_zero(float* __restrict__ p, int n4) {
  int i = blockIdx.x * blockDim.x + threadIdx.x;
  if (i < n4) ((float4*)p)[i] = make_float4(0.f, 0.f, 0.f, 0.f);
}

// h = x @ W_in2f^T + b_in2f   [nAtoms x 128]
__global__ void __launch_bounds__(256) k_in2f(const float* __restrict__ x,
                                              const float* __restrict__ W,
                                              const float* __restrict__ b,
                                              float* __restrict__ h, int nAtoms) {
  __shared__ float xs[16 * PITCH];
  const int r0 = blockIdx.x * 16;
  for (int i = threadIdx.x; i < 16 * NF; i += 256) {
    int m = i >> 7, k = i & (NF - 1);
    int r = min(r0 + m, nAtoms - 1);
    xs[m * PITCH + k] = x[(size_t)r * NF + k];
  }
  __syncthreads();
  const int lane = threadIdx.x & 31, wave = threadIdx.x >> 5;
  const int nb = wave * 16, l16 = lane & 15, half = lane >> 4;
  v8f c = {};
  c = wmma_f32_acc<NF, PITCH>(xs, W, NF, nb, c, lane);
  const float bias = b[nb + l16];
#pragma unroll
  for (int v = 0; v < 8; ++v) {
    int m = r0 + v + 8 * half;
    if (m < nAtoms) h[(size_t)m * NF + nb + l16] = c[v] + bias;
  }
}

// Fused edge pipeline over 64 edges/block, weights register-resident:
//   Wij = (ssp(f_ij@W_f1^T+b1))@W_f2^T+b2;  Wij *= rcut;
//   agg[idx_i] += h[idx_j] * Wij            (global f32 atomics, L2-resident)
__global__ void __launch_bounds__(256) k_edges(const float* __restrict__ f_ij,
                                               const int* __restrict__ idx_i,
                                               const int* __restrict__ idx_j,
                                               const float* __restrict__ rcut,
                                               const float* __restrict__ Wf1,
                                               const float* __restrict__ bf1,
                                               const float* __restrict__ Wf2,
                                               const float* __restrict__ bf2,
                                               const float* __restrict__ h,
                                               float* __restrict__ agg,
                                               int nEdges) {
  __shared__ float fs[SUBT][16 * NRBF];
  __shared__ float ts[SUBT][16 * PITCH];
  __shared__ int   sii[SUBT * 16];
  __shared__ int   sjj[SUBT * 16];
  __shared__ float src[SUBT * 16];
  const int e0 = blockIdx.x * (SUBT * 16);

  // stage f_ij rows for all 64 edges (coalesced)
  for (int i = threadIdx.x; i < SUBT * 16 * NRBF; i += 256) {
    int m = i / NRBF, k = i - m * NRBF;
    int e = min(e0 + m, nEdges - 1);
    fs[m >> 4][(m & 15) * NRBF + k] = f_ij[(size_t)e * NRBF + k];
  }
  if (threadIdx.x < SUBT * 16) {
    int e = min(e0 + (int)threadIdx.x, nEdges - 1);
    sii[threadIdx.x] = idx_i[e];
    sjj[threadIdx.x] = idx_j[e];
    src[threadIdx.x] = rcut[e];
  }
  __syncthreads();

  const int lane = threadIdx.x & 31, wave = threadIdx.x >> 5;
  const int nb = wave * 16, l16 = lane & 15, half = lane >> 4;
  const int koff = half << 1;

  // Pre-load edge-invariant B fragments into registers.
  v2f B1[NRBF / 4];                       // W_f1 chunk: 5 k-steps
  {
    const float* w1 = Wf1 + (size_t)(nb + l16) * NRBF + koff;
#pragma unroll
    for (int k = 0; k < NRBF / 4; ++k) { B1[k].x = w1[4 * k]; B1[k].y = w1[4 * k + 1]; }
  }
  v2f B2[NF / 4];                         // W_f2 chunk: 32 k-steps
  {
    const float* w2 = Wf2 + (size_t)(nb + l16) * NF + koff;
#pragma unroll
    for (int k = 0; k < NF / 4; ++k) { B2[k].x = w2[4 * k]; B2[k].y = w2[4 * k + 1]; }
  }
  const float b1 = bf1[nb + l16];
  const float b2 = bf2[nb + l16];

  // Phase 1: GEMM1 (K=20) + shifted softplus -> ts  (B2 loads overlap these WMMAs)
#pragma unroll
  for (int st = 0; st < SUBT; ++st) {
    v8f c = {};
    c = wmma_f32_accB<NRBF / 4, NRBF>(fs[st], B1, c, lane);
#pragma unroll
    for (int v = 0; v < 8; ++v)
      ts[st][(v + 8 * half) * PITCH + nb + l16] = ssp(c[v] + b1);
  }
  __syncthreads();

  // Phase 2: GEMM2 (K=128) + cutoff + gather/scatter
#pragma unroll
  for (int st = 0; st < SUBT; ++st) {
    v8f c2 = {};
    c2 = wmma_f32_accB<NF / 4, PITCH>(ts[st], B2, c2, lane);
#pragma unroll
    for (int v = 0; v < 8; ++v) {
      int m = st * 16 + v + 8 * half;
      int e = e0 + m;
      if (e < nEdges) {
        float wij = (c2[v] + b2) * src[m];
        float hv  = h[(size_t)sjj[m] * NF + nb + l16];            // gather (L2)
        atomicAdd(&agg[(size_t)sii[m] * NF + nb + l16], wij * hv); // scatter (L2)
      }
    }
  }
}

// out = ssp(agg @ W_o1^T + b_o1) @ W_o2^T + b_o2
__global__ void __launch_bounds__(256) k_out(const float* __restrict__ agg,
                                             const float* __restrict__ Wo1,
                                             const float* __restrict__ bo1,
                                             const float* __restrict__ Wo2,
                                             const float* __restrict__ bo2,
                                             float* __restrict__ out, int nAtoms) {
  __shared__ float as[16 * PITCH];
  __shared__ float ts[16 * PITCH];
  const int r0 = blockIdx.x * 16;
  for (int i = threadIdx.x; i < 16 * NF; i += 256) {
    int m = i >> 7, k = i & (NF - 1);
    int r = min(r0 + m, nAtoms - 1);
    as[m * PITCH + k] = agg[(size_t)r * NF + k];
  }
  __syncthreads();
  const int lane = threadIdx.x & 31, wave = threadIdx.x >> 5;
  const int nb = wave * 16, l16 = lane & 15, half = lane >> 4;

  v8f c = {};
  c = wmma_f32_acc<NF, PITCH>(as, Wo1, NF, nb, c, lane);
  const float b1 = bo1[nb + l16];
#pragma unroll
  for (int v = 0; v < 8; ++v)
    ts[(v + 8 * half) * PITCH + nb + l16] = ssp(c[v] + b1);
  __syncthreads();

  v8f c2 = {};
  c2 = wmma_f32_acc<NF, PITCH>(ts, Wo2, NF, nb, c2, lane);
  const float b2 = bo2[nb + l16];
#pragma unroll
  for (int v = 0; v < 8; ++v) {
    int m = r0 + v + 8 * half;
    if (m < nAtoms) out[(size_t)m * NF + nb + l16] = c2[v] + b2;
  }
}

extern "C" void kernel_launch(void* const* d_in, const int* in_sizes, int n_in,
                              void* d_out, int out_size, void* d_ws, size_t ws_size,
                              hipStream_t stream) {
  const float* x      = (const float*)d_in[0];
  const float* f_ij   = (const float*)d_in[1];
  const int*   idx_i  = (const int*)d_in[2];
  const int*   idx_j  = (const int*)d_in[3];
  const float* rcut   = (const float*)d_in[4];
  const float* W_in2f = (const float*)d_in[5];
  const float* b_in2f = (const float*)d_in[6];
  const float* W_f1   = (const float*)d_in[7];
  const float* b_f1   = (const float*)d_in[8];
  const float* W_f2   = (const float*)d_in[9];
  const float* b_f2   = (const float*)d_in[10];
  const float* W_o1   = (const float*)d_in[11];
  const float* b_o1   = (const float*)d_in[12];
  const float* W_o2   = (const float*)d_in[13];
  const float* b_o2   = (const float*)d_in[14];

  const int nAtoms = in_sizes[0] / NF;
  const int nEdges = in_sizes[2];

  float* h   = (float*)d_ws;                       // [nAtoms * 128]
  float* agg = h + (size_t)nAtoms * NF;            // [nAtoms * 128]

  const int atomTiles  = (nAtoms + 15) / 16;
  const int edgeBlocks = (nEdges + SUBT * 16 - 1) / (SUBT * 16);
  const int n4 = (nAtoms * NF) / 4;

  k_zero <<<(n4 + 255) / 256, 256, 0, stream>>>(agg, n4);
  k_in2f <<<atomTiles, 256, 0, stream>>>(x, W_in2f, b_in2f, h, nAtoms);
  k_edges<<<edgeBlocks, 256, 0, stream>>>(f_ij, idx_i, idx_j, rcut,
                                          W_f1, b_f1, W_f2, b_f2, h, agg, nEdges);
  k_out  <<<atomTiles, 256, 0, stream>>>(agg, W_o1, b_o1, W_o2, b_o2,
                                         (float*)d_out, nAtoms);
}